// GNNEncoder_49031346651816
// MI455X (gfx1250) — compile-verified
//
#include <hip/hip_runtime.h>
#include <hip/hip_bf16.h>

typedef __attribute__((ext_vector_type(2))) float v2f;
typedef __attribute__((ext_vector_type(8))) float v8f;
typedef __attribute__((ext_vector_type(4))) unsigned int v4u;
typedef __attribute__((ext_vector_type(8))) int v8i;
typedef __attribute__((ext_vector_type(4))) int v4i;

#define F 128
#define BN_EPS 1e-5f

#if defined(__has_include)
#if __has_include(<hip/amd_detail/amd_gfx1250_TDM.h>)
#define TDM_6ARG 1
#else
#define TDM_6ARG 0
#endif
#else
#define TDM_6ARG 0
#endif

// ---------------------------------------------------------------------------
// Utility: zero a float buffer
// ---------------------------------------------------------------------------
__global__ void zerof(float* __restrict__ p, int n) {
  int i = blockIdx.x * blockDim.x + threadIdx.x;
  if (i < n) p[i] = 0.0f;
}

// ---------------------------------------------------------------------------
// Degree: deg[dst] += 1 over edges, then dinv_sqrt / inv_deg (with self-loop)
// ---------------------------------------------------------------------------
__global__ void deg_count(float* __restrict__ deg, const int* __restrict__ ei, int E) {
  int e = blockIdx.x * blockDim.x + threadIdx.x;
  if (e < E) unsafeAtomicAdd(&deg[ei[E + e]], 1.0f);
}

__global__ void deg_final(const float* __restrict__ deg, float* __restrict__ dis,
                          float* __restrict__ idg, int N) {
  int i = blockIdx.x * blockDim.x + threadIdx.x;
  if (i < N) {
    float d = deg[i] + 1.0f;  // self-loop
    dis[i] = rsqrtf(d);
    idg[i] = 1.0f / d;
  }
}

// ---------------------------------------------------------------------------
// Stage W[128x128] f32 into LDS via the Tensor Data Mover (one DMA per block,
// issued by wave 0). D# per CDNA5 ISA:
//   group0: count=1 | lds_addr | global_addr | type=2
//   group1: data_size=4B, tensor_dim0/1 = 128, tile_dim0/1 = 128, stride0 = 128
//   groups 2/3: zero (2D tensor)
// ---------------------------------------------------------------------------
__device__ __forceinline__ void stage_W_tdm(float* sW, const float* W, int tid) {
#if __has_builtin(__builtin_amdgcn_tensor_load_to_lds) && \
    __has_builtin(__builtin_amdgcn_s_wait_tensorcnt)
  if (tid < 32) {  // wave 0 issues the DMA (TDM ignores EXEC; one wave only)
    unsigned lds_off = (unsigned)(unsigned long long)(uintptr_t)sW;
    unsigned long long ga = (unsigned long long)(uintptr_t)W;
    v4u g0;
    g0[0] = 1u;                                  // count=1, no gather
    g0[1] = lds_off;                             // LDS byte address
    g0[2] = (unsigned)(ga & 0xFFFFFFFFull);      // global_addr[31:0]
    g0[3] = (unsigned)((ga >> 32) & 0x01FFFFFFull) | 0x80000000u;  // [56:32]|type=2
    v8i g1;
    g1[0] = 0x00020000;        // data_size=2 (4 bytes); mask/flags/pad = 0
    g1[1] = (int)(128u << 16); // tensor_dim0[15:0] in bits[63:48]
    g1[2] = (int)(128u << 16); // tensor_dim0 hi=0; tensor_dim1[15:0] in [95:80]
    g1[3] = (int)(128u << 16); // tensor_dim1 hi=0; tile_dim0=128 in [127:112]
    g1[4] = 128;               // tile_dim1=128; tile_dim2=0
    g1[5] = 128;               // tensor_dim0_stride = 128 elements
    g1[6] = 0;
    g1[7] = 0;
    v4i gz = {0, 0, 0, 0};
#if TDM_6ARG
    v8i gz8 = {0, 0, 0, 0, 0, 0, 0, 0};
    __builtin_amdgcn_tensor_load_to_lds(g0, g1, gz, gz, gz8, 0);
#else
    __builtin_amdgcn_tensor_load_to_lds(g0, g1, gz, gz, 0);
#endif
    __builtin_amdgcn_s_wait_tensorcnt(0);
  }
  __syncthreads();
#else
  for (int i = tid; i < F * F; i += 256) sW[i] = W[i];
  __syncthreads();
#endif
}

// ---------------------------------------------------------------------------
// GEMM + fused self-loop init:
//   T[N x 128] = X @ W          (fp32 WMMA 16x16x4)
//   H[N x 128] = inv_deg .* T   (aggregation accumulator, pre-seeded)
// Block = 256 threads (8 waves), 128 rows per block, W staged in LDS by TDM.
// A layout (16x4 f32): lanes 0-15 -> M=lane, K = 4s + {0,1};
//                      lanes 16-31 -> M=lane-16, K = 4s + {2,3}.
// B layout mirrored: lane%16 = N column, K = 4s + 2*(lane/16) + vgpr.
// C/D: vgpr r -> M = r + 8*(lane/16), N = lane%16.
// ---------------------------------------------------------------------------
__global__ __launch_bounds__(256) void gemm128_wmma(const float* __restrict__ X,
                                                    const float* __restrict__ W,
                                                    float* __restrict__ T,
                                                    float* __restrict__ H,
                                                    const float* __restrict__ idg,
                                                    int N) {
  __shared__ float sW[F * F];
  const int t = threadIdx.x;
  stage_W_tdm(sW, W, t);

  const int wave = t >> 5;
  const int lane = t & 31;
  const int h = lane >> 4;   // lane half: 0 or 1
  const int n = lane & 15;   // tile column / A row
  const int rowBase = blockIdx.x * 128 + wave * 16;
  // Rows >= N are clamped: their products are computed but never stored,
  // keeping the inner loop free of EXEC manipulation.
  const int arow = min(rowBase + n, N - 1);
  const float* __restrict__ xrow = X + (size_t)arow * F;

  v8f acc[8];
#pragma unroll
  for (int j = 0; j < 8; ++j)
#pragma unroll
    for (int r = 0; r < 8; ++r) acc[j][r] = 0.0f;

  for (int s = 0; s < 32; ++s) {
    const int k0 = s * 4 + h * 2;
    v2f a = *reinterpret_cast<const v2f*>(xrow + k0);  // global_load_b64
#pragma unroll
    for (int j = 0; j < 8; ++j) {
      v2f b;
      b[0] = sW[k0 * F + j * 16 + n];
      b[1] = sW[(k0 + 1) * F + j * 16 + n];
      acc[j] = __builtin_amdgcn_wmma_f32_16x16x4_f32(
          /*neg_a=*/false, a, /*neg_b=*/false, b,
          /*c_mod=*/(short)0, acc[j], /*reuse_a=*/false, /*reuse_b=*/false);
    }
  }

#pragma unroll
  for (int r = 0; r < 8; ++r) {
    const int row = rowBase + h * 8 + r;
    if (row < N) {
      const float scale = idg[row];
      const size_t base = (size_t)row * F + n;
#pragma unroll
      for (int j = 0; j < 8; ++j) {
        const float v = acc[j][r];
        T[base + j * 16] = v;
        H[base + j * 16] = scale * v;  // self-loop term pre-added
      }
    }
  }
}

// ---------------------------------------------------------------------------
// Edge scatter: one wave per edge; lane handles 4 features (float4 gather,
// 4x native f32 atomic add into H[dst]). 51MB feature matrix lives in L2.
// ---------------------------------------------------------------------------
__global__ __launch_bounds__(256) void edge_scatter(float* __restrict__ H,
                                                    const float* __restrict__ T,
                                                    const int* __restrict__ ei,
                                                    const float* __restrict__ dis,
                                                    int E) {
  int e = blockIdx.x * 8 + (threadIdx.x >> 5);
  if (e >= E) return;
  int lane = threadIdx.x & 31;
  int src = ei[e];
  int dst = ei[E + e];
  float norm = dis[src] * dis[dst];
  const float4 v = *reinterpret_cast<const float4*>(T + (size_t)src * F + lane * 4);
  float* hp = H + (size_t)dst * F + lane * 4;
  unsafeAtomicAdd(hp + 0, norm * v.x);
  unsafeAtomicAdd(hp + 1, norm * v.y);
  unsafeAtomicAdd(hp + 2, norm * v.z);
  unsafeAtomicAdd(hp + 3, norm * v.w);
}

// ---------------------------------------------------------------------------
// Combine: out = [relu]( gamma * ((H + bias) * rsqrt(1+eps)) + beta )
// ---------------------------------------------------------------------------
__global__ void combine(float* __restrict__ out, const float* __restrict__ H,
                        const float* __restrict__ b, const float* __restrict__ g,
                        const float* __restrict__ be, int NF, int do_relu) {
  int idx = blockIdx.x * blockDim.x + threadIdx.x;
  if (idx < NF) {
    int f = idx & (F - 1);
    float v = (H[idx] + b[f]) * rsqrtf(1.0f + BN_EPS);
    v = g[f] * v + be[f];
    if (do_relu) v = fmaxf(v, 0.0f);
    out[idx] = v;
  }
}

// ---------------------------------------------------------------------------
// Pooling
// ---------------------------------------------------------------------------
__global__ void pool_cnt(float* __restrict__ cnt, const int* __restrict__ batch, int N) {
  int i = blockIdx.x * blockDim.x + threadIdx.x;
  if (i < N) unsafeAtomicAdd(&cnt[batch[i]], 1.0f);
}

__global__ void pool_sum(float* __restrict__ sums, const float* __restrict__ ne,
                         const int* __restrict__ batch, int N) {
  int tid = blockIdx.x * blockDim.x + threadIdx.x;
  int i = tid >> 5;
  int q = tid & 31;
  if (i < N) {
    int gph = batch[i];
    const float4 v = *reinterpret_cast<const float4*>(ne + (size_t)i * F + q * 4);
    float* sp = sums + (size_t)gph * F + q * 4;
    unsafeAtomicAdd(sp + 0, v.x);
    unsafeAtomicAdd(sp + 1, v.y);
    unsafeAtomicAdd(sp + 2, v.z);
    unsafeAtomicAdd(sp + 3, v.w);
  }
}

// ---------------------------------------------------------------------------
// MLP head (G=64 rows; compute-trivial, simple scalar kernels)
// ---------------------------------------------------------------------------
__global__ void mlp1(float* __restrict__ tmp, const float* __restrict__ sums,
                     const float* __restrict__ cnt, const float* __restrict__ P1,
                     const float* __restrict__ pb1, int GF) {
  int idx = blockIdx.x * blockDim.x + threadIdx.x;
  if (idx < GF) {
    int gph = idx >> 7;
    int o = idx & (F - 1);
    float invc = 1.0f / fmaxf(cnt[gph], 1.0f);
    const float* srow = sums + (size_t)gph * F;
    float acc = 0.0f;
    for (int k = 0; k < F; ++k) acc += (srow[k] * invc) * P1[k * F + o];
    tmp[idx] = fmaxf(acc + pb1[o], 0.0f);
  }
}

__global__ void mlp2(float* __restrict__ out, const float* __restrict__ tmp,
                     const float* __restrict__ P2, const float* __restrict__ pb2, int GF) {
  int idx = blockIdx.x * blockDim.x + threadIdx.x;
  if (idx < GF) {
    int gph = idx >> 7;
    int o = idx & (F - 1);
    const float* trow = tmp + (size_t)gph * F;
    float acc = 0.0f;
    for (int k = 0; k < F; ++k) acc += trow[k] * P2[k * F + o];
    out[idx] = acc + pb2[o];
  }
}

// ---------------------------------------------------------------------------
// Launch
// ---------------------------------------------------------------------------
extern "C" void kernel_launch(void* const* d_in, const int* in_sizes, int n_in,
                              void* d_out, int out_size, void* d_ws, size_t ws_size,
                              hipStream_t stream) {
  const float* x = (const float*)d_in[0];
  const int* ei = (const int*)d_in[1];
  const int* batch = (const int*)d_in[2];
  const float* W1 = (const float*)d_in[3];
  const float* b1 = (const float*)d_in[4];
  const float* W2 = (const float*)d_in[5];
  const float* b2 = (const float*)d_in[6];
  const float* W3 = (const float*)d_in[7];
  const float* b3 = (const float*)d_in[8];
  const float* g1 = (const float*)d_in[9];
  const float* be1 = (const float*)d_in[10];
  const float* g2 = (const float*)d_in[11];
  const float* be2 = (const float*)d_in[12];
  const float* g3 = (const float*)d_in[13];
  const float* be3 = (const float*)d_in[14];
  const float* P1 = (const float*)d_in[15];
  const float* pb1 = (const float*)d_in[16];
  const float* P2 = (const float*)d_in[17];
  const float* pb2 = (const float*)d_in[18];

  const int N = in_sizes[0] / F;
  const int E = in_sizes[1] / 2;
  const int G = (out_size - N * F) / F;
  const int NF = N * F;
  const int GF = G * F;

  float* ws = (float*)d_ws;
  float* T = ws;                          // N*F  gemm output
  float* H = T + (size_t)NF;              // N*F  aggregation accumulator
  float* deg = H + (size_t)NF;            // N
  float* dis = deg + N;                   // N    dinv_sqrt
  float* idg = dis + N;                   // N    inv_deg
  float* sums = idg + N;                  // G*F
  float* cnt = sums + (size_t)GF;         // G
  float* tmp = cnt + G;                   // G*F

  float* out_nodes = (float*)d_out;
  float* out_graph = out_nodes + (size_t)NF;

  auto cdiv = [](int a, int b) { return (a + b - 1) / b; };
  dim3 blk(256);

  // degrees
  zerof<<<cdiv(N, 256), blk, 0, stream>>>(deg, N);
  deg_count<<<cdiv(E, 256), blk, 0, stream>>>(deg, ei, E);
  deg_final<<<cdiv(N, 256), blk, 0, stream>>>(deg, dis, idg, N);

  const float* Ws[3] = {W1, W2, W3};
  const float* bs[3] = {b1, b2, b3};
  const float* gs[3] = {g1, g2, g3};
  const float* bes[3] = {be1, be2, be3};

  const float* Xin = x;
  for (int layer = 0; layer < 3; ++layer) {
    gemm128_wmma<<<cdiv(N, 128), blk, 0, stream>>>(Xin, Ws[layer], T, H, idg, N);
    edge_scatter<<<cdiv(E, 8), blk, 0, stream>>>(H, T, ei, dis, E);
    float* dstp = (layer == 2) ? out_nodes : H;
    combine<<<cdiv(NF, 256), blk, 0, stream>>>(dstp, H, bs[layer], gs[layer],
                                               bes[layer], NF, layer != 2);
    Xin = H;
  }

  // pooling (sums and cnt are adjacent -> single zero pass)
  zerof<<<cdiv(GF + G, 256), blk, 0, stream>>>(sums, GF + G);
  pool_cnt<<<cdiv(N, 256), blk, 0, stream>>>(cnt, batch, N);
  pool_sum<<<cdiv(N * 32, 256), blk, 0, stream>>>(sums, out_nodes, batch, N);

  // MLP head
  mlp1<<<cdiv(GF, 256), blk, 0, stream>>>(tmp, sums, cnt, P1, pb1, GF);
  mlp2<<<cdiv(GF, 256), blk, 0, stream>>>(out_graph, tmp, P2, pb2, GF);
}